// GIN_61950608277614
// MI455X (gfx1250) — compile-verified
//
#include <hip/hip_runtime.h>

typedef float v2f __attribute__((ext_vector_type(2)));
typedef float v8f __attribute__((ext_vector_type(8)));

#define TM 64          // rows (nodes) per block
#define KC 32          // K chunk staged in LDS
#define LDA (KC + 2)   // even padded stride: 8B-aligned pairs + conflict-free

// ---------------------------------------------------------------------------
// Templated fp32 WMMA GEMM: out[N x NOUT] = act(A[N x K] @ W[K x NOUT] + bias)
// A staged row-major [TM][LDA]; W staged TRANSPOSED [NOUT][LDA] so each lane's
// B fragment {W[k][col], W[k+1][col]} is one aligned ds_load_b64 (no repack).
// 256 threads = 8 waves; wave (m,n) owns a 16-row slab x NOUT/2 col slab.
// ---------------------------------------------------------------------------
template <int K, int NOUT, int RELU>
__global__ __launch_bounds__(256) void gin_gemm_kernel(
    const float* __restrict__ A, const float* __restrict__ W,
    const float* __restrict__ bias, float* __restrict__ out, int N)
{
    constexpr int HALFN  = NOUT / 2;      // 64 or 128
    constexpr int NTILES = HALFN / 16;    // 4 or 8

    __shared__ float As[TM][LDA];         // 64 x 32 A chunk (row-major)
    __shared__ float Wt[NOUT][LDA];       // 32 x NOUT W chunk, transposed

    const int tid   = threadIdx.x;
    const int lane  = tid & 31;
    const int wave  = tid >> 5;           // 0..7
    const int waveM = wave >> 1;          // 0..3 -> 16-row slab
    const int waveN = wave & 1;           // 0..1 -> half of NOUT
    const int half  = lane >> 4;          // 0/1 (K-pair select per ISA A layout)
    const int m     = lane & 15;          // row within 16 (A) / col within 16 (B)

    const int rowBase = blockIdx.x * TM;
    const int colBase = waveN * HALFN;

    v8f acc[NTILES];
    #pragma unroll
    for (int i = 0; i < NTILES; ++i) acc[i] = (v8f){};

    for (int kk = 0; kk < K; kk += KC) {
        // Stage A chunk (global reads coalesced along K)
        #pragma unroll
        for (int e = tid; e < TM * KC; e += 256) {
            int r = e >> 5;               // e / KC
            int c = e & (KC - 1);         // e % KC
            int row = rowBase + r;
            As[r][c] = (row < N) ? A[(long)row * K + (kk + c)] : 0.0f;
        }
        // Stage W chunk transposed (global reads coalesced along NOUT;
        // LDS writes stride-34 across lanes -> all 32 banks distinct)
        #pragma unroll
        for (int e = tid; e < KC * NOUT; e += 256) {
            int r = e / NOUT;
            int c = e - r * NOUT;
            Wt[c][r] = W[(long)(kk + r) * NOUT + c];
        }
        __syncthreads();

        // 8 K-steps of 4 per chunk; EXEC is all ones at every WMMA
        #pragma unroll
        for (int k2 = 0; k2 < KC; k2 += 4) {
            // A fragment 16x4: lanes 0-15 hold K={0,1}, lanes 16-31 K={2,3};
            // contiguous aligned pair -> single ds_load_b64
            const int ar = 16 * waveM + m;
            const int ak = k2 + 2 * half;          // even
            const v2f a = *(const v2f*)&As[ar][ak];
            #pragma unroll
            for (int nt = 0; nt < NTILES; ++nt) {
                // B fragment 4x16 from transposed tile: contiguous aligned pair
                const int col = colBase + nt * 16 + m;
                const v2f b = *(const v2f*)&Wt[col][ak];
                acc[nt] = __builtin_amdgcn_wmma_f32_16x16x4_f32(
                    /*neg_a=*/false, a, /*neg_b=*/false, b,
                    /*c_mod=*/(short)0, acc[nt],
                    /*reuse_a=*/false, /*reuse_b=*/false);
            }
        }
        __syncthreads();
    }

    // Epilogue: C/D layout — VGPR r: lanes0-15 -> M=r, lanes16-31 -> M=8+r
    #pragma unroll
    for (int nt = 0; nt < NTILES; ++nt) {
        const int col = colBase + nt * 16 + m;
        const float bv = bias ? bias[col] : 0.0f;
        #pragma unroll
        for (int r = 0; r < 8; ++r) {
            const int row = rowBase + 16 * waveM + 8 * half + r;
            if (row < N) {
                float v = acc[nt][r] + bv;
                if (RELU) v = fmaxf(v, 0.0f);
                out[(long)row * NOUT + col] = v;
            }
        }
    }
}

// ---------------------------------------------------------------------------
// z = (1 + eps) * h   (also fully initializes z, overwriting workspace poison)
// ---------------------------------------------------------------------------
__global__ __launch_bounds__(256) void gin_scale_kernel(
    const float* __restrict__ h, const float* __restrict__ eps,
    float* __restrict__ z, long n4)
{
    long i = (long)blockIdx.x * blockDim.x + threadIdx.x;
    if (i >= n4) return;
    const float s = 1.0f + eps[0];
    float4 v = ((const float4*)h)[i];
    v.x *= s; v.y *= s; v.z *= s; v.w *= s;
    ((float4*)z)[i] = v;
}

// ---------------------------------------------------------------------------
// Edge scatter: z[dst] += h[src]. One edge per wave32; each lane carries 4
// contiguous features (float4 gather + 4 f32 atomics; the 51 MB target array
// is L2-resident on MI455X's 192 MB L2, so atomics stay on-chip).
// ---------------------------------------------------------------------------
__global__ __launch_bounds__(256) void gin_edge_scatter_kernel(
    const float* __restrict__ h, const int* __restrict__ src,
    const int* __restrict__ dst, float* __restrict__ z, long nE)
{
    const long e = (long)blockIdx.x * 8 + (threadIdx.x >> 5);
    if (e >= nE) return;
    const int lane = threadIdx.x & 31;
    const long s = src[e];
    const long d = dst[e];
    const float4 v = ((const float4*)(h + s * 128))[lane];
    float* zd = z + d * 128 + lane * 4;
    atomicAdd(zd + 0, v.x);
    atomicAdd(zd + 1, v.y);
    atomicAdd(zd + 2, v.z);
    atomicAdd(zd + 3, v.w);
}

// ---------------------------------------------------------------------------
extern "C" void kernel_launch(void* const* d_in, const int* in_sizes, int n_in,
                              void* d_out, int out_size, void* d_ws, size_t ws_size,
                              hipStream_t stream) {
    const float* x       = (const float*)d_in[0];
    const int*   eidx    = (const int*)  d_in[1];
    const float* W_embed = (const float*)d_in[2];
    const float* eps1    = (const float*)d_in[3];
    const float* w1a     = (const float*)d_in[4];
    const float* b1a     = (const float*)d_in[5];
    const float* w1b     = (const float*)d_in[6];
    const float* b1b     = (const float*)d_in[7];
    const float* eps2    = (const float*)d_in[8];
    const float* w2a     = (const float*)d_in[9];
    const float* b2a     = (const float*)d_in[10];
    const float* w2b     = (const float*)d_in[11];
    const float* b2b     = (const float*)d_in[12];
    const float* eps3    = (const float*)d_in[13];
    const float* w3a     = (const float*)d_in[14];
    const float* b3a     = (const float*)d_in[15];
    const float* w3b     = (const float*)d_in[16];
    const float* b3b     = (const float*)d_in[17];

    const int  N = in_sizes[0] / 64;       // 100000
    const long E = in_sizes[1] / 2;        // 1600000
    const int* src = eidx;
    const int* dst = eidx + E;

    // Buffers: h lives in d_out (N x 128); z, t in workspace.
    float* h = (float*)d_out;                       // N x 128
    float* z = (float*)d_ws;                        // N x 128
    float* t = (float*)d_ws + (long)N * 128;        // N x 256

    const dim3 blk(256);
    const dim3 gemmGrid((N + TM - 1) / TM);
    const long n4 = (long)N * 128 / 4;
    const dim3 scaleGrid((unsigned)((n4 + 255) / 256));
    const dim3 edgeGrid((unsigned)((E + 7) / 8));

    // h = x @ W_embed (no bias)
    gin_gemm_kernel<64, 128, 0><<<gemmGrid, blk, 0, stream>>>(x, W_embed, nullptr, h, N);

    // ----- conv1 -----
    gin_scale_kernel<<<scaleGrid, blk, 0, stream>>>(h, eps1, z, n4);
    gin_edge_scatter_kernel<<<edgeGrid, blk, 0, stream>>>(h, src, dst, z, E);
    gin_gemm_kernel<128, 128, 1><<<gemmGrid, blk, 0, stream>>>(z, w1a, b1a, t, N);
    gin_gemm_kernel<128, 128, 0><<<gemmGrid, blk, 0, stream>>>(t, w1b, b1b, h, N);

    // ----- conv2 -----
    gin_scale_kernel<<<scaleGrid, blk, 0, stream>>>(h, eps2, z, n4);
    gin_edge_scatter_kernel<<<edgeGrid, blk, 0, stream>>>(h, src, dst, z, E);
    gin_gemm_kernel<128, 256, 1><<<gemmGrid, blk, 0, stream>>>(z, w2a, b2a, t, N);
    gin_gemm_kernel<256, 128, 1><<<gemmGrid, blk, 0, stream>>>(t, w2b, b2b, h, N);

    // ----- conv3 -----
    gin_scale_kernel<<<scaleGrid, blk, 0, stream>>>(h, eps3, z, n4);
    gin_edge_scatter_kernel<<<edgeGrid, blk, 0, stream>>>(h, src, dst, z, E);
    gin_gemm_kernel<128, 256, 1><<<gemmGrid, blk, 0, stream>>>(z, w3a, b3a, t, N);
    gin_gemm_kernel<256, 128, 1><<<gemmGrid, blk, 0, stream>>>(t, w3b, b3b, h, N);
}